// MultiPropGNN_48988396978373
// MI455X (gfx1250) — compile-verified
//
#include <hip/hip_runtime.h>
#include <hip/hip_bf16.h>
#include <math.h>

// MI455X / gfx1250, wave32. Heavy per-edge GEMMs on V_WMMA_F32_16X16X4_F32.

typedef __attribute__((ext_vector_type(2))) float v2f;
typedef __attribute__((ext_vector_type(8))) float v8f;

#define LDIM 112   // LABEL_DIM
#define LEMB 8     // LABEL_EMB
#define NLK  16    // LABEL_K

static __device__ __forceinline__ v8f v8zero() {
  v8f z = {0.f,0.f,0.f,0.f,0.f,0.f,0.f,0.f};
  return z;
}

static __device__ __forceinline__ v8f wmma4(v2f a, v2f b, v8f c) {
  // D = A(16x4,f32) * B(4x16,f32) + C(16x16,f32)
  return __builtin_amdgcn_wmma_f32_16x16x4_f32(false, a, false, b, (short)0, c, false, false);
}

// Wave-local LDS fence: same-wave cross-lane LDS RAW (DS ops are in-order per
// wave; we just need the wait + a compiler reordering barrier).
static __device__ __forceinline__ void wave_lds_fence() {
  __builtin_amdgcn_wave_barrier();
  asm volatile("s_wait_dscnt 0" ::: "memory");
  __builtin_amdgcn_wave_barrier();
}

// ---------------------------------------------------------------------------
// Heavy per-edge kernel. Block = 128 threads = 4 wave32; each wave owns a tile
// of 16 edges (grid-stride). Weights staged once per block into LDS.
// ---------------------------------------------------------------------------
template<int C>
__launch_bounds__(128, 1)
__global__ void edge_kernel(
    const float* __restrict__ y, const int* __restrict__ ei,
    const float* __restrict__ eattr, const int* __restrict__ emask,
    const float* __restrict__ q,
    const float* __restrict__ table,
    const float* __restrict__ Wi2k, const float* __restrict__ bi2k,
    const float* __restrict__ We2o, const float* __restrict__ be2o,
    const float* __restrict__ Wkkey, const float* __restrict__ bkkey,
    const float* __restrict__ Wkedge, const float* __restrict__ bkedge,
    const float* __restrict__ Walpha, const float* __restrict__ balpha,
    float* __restrict__ out_edge, float* __restrict__ aedge,
    unsigned int* __restrict__ amaxenc, int E)
{
  constexpr int Ct    = C / 16;
  constexpr int TW_LD = 136;      // 2*136 % 64 == 16 -> halves hit disjoint banks
  constexpr int KK_LD = C + 8;    // 2*(C+8) % 64 in {16,48} for C in {64,112}
  constexpr int LK_LD = 116;      // 8*116 % 64 == 32
  constexpr int KP_LD = 132;      // 8*132 % 64 == 32
  constexpr int PW    = 16*LK_LD + 16*8 + 16*KP_LD + 48;   // per-wave floats

  extern __shared__ float smem[];
  float* sTW = smem;                    // [112][TW_LD]: TW[d][m*16+k]=table[d][m]*Wi2k[k][d]
  float* sKK = sTW + LDIM*TW_LD;        // [C][KK_LD]:  B[c'][c] = Wkkey[c][c']

  const int wave = threadIdx.x >> 5;
  const int lane = threadIdx.x & 31;
  const int nl   = lane & 15;
  const int hi   = lane >> 4;

  float* sW   = sKK + C*KK_LD + wave*PW;
  float* sLK  = sW;                     // [16][LK_LD]: L tile, later k_labels stage
  float* sEA  = sLK + 16*LK_LD;         // [16][8] edge_attr
  float* sKP  = sEA + 16*8;             // [16][KP_LD] kpre (j = m*16+k)
  int*   sSrc = (int*)(sKP + 16*KP_LD); // [16]
  int*   sDst = sSrc + 16;              // [16]
  float* sMsk = (float*)(sDst + 16);    // [16]

  for (int i = threadIdx.x; i < LDIM*128; i += blockDim.x) {
    int d = i >> 7, j = i & 127, m = j >> 4, k = j & 15;
    sTW[d*TW_LD + j] = table[d*LEMB + m] * Wi2k[k*LDIM + d];
  }
  for (int i = threadIdx.x; i < C*C; i += blockDim.x) {
    int cp = i / C, c = i - cp*C;
    sKK[cp*KK_LD + c] = Wkkey[c*C + cp];
  }
  __syncthreads();

  const int nTiles = (E + 15) >> 4;
  for (int tile = blockIdx.x*4 + wave; tile < nTiles; tile += gridDim.x*4) {
    const int e0 = tile << 4;
    if (lane < 16) {
      int e = min(e0 + lane, E - 1);
      int s = ei[e];
      sSrc[lane] = s;
      sDst[lane] = ei[E + e];
      sMsk[lane] = (float)emask[s];
    }
    wave_lds_fence();
    // L tile: y[src] rows [16][112]
    for (int i = lane; i < 16*LDIM; i += 32) {
      int ee = i / LDIM, d = i - ee*LDIM;
      sLK[ee*LK_LD + d] = y[(size_t)sSrc[ee]*LDIM + d];
    }
    for (int i = lane; i < 16*8; i += 32) {
      int ee = i >> 3, j = i & 7;
      int e = min(e0 + ee, E - 1);
      sEA[ee*8 + j] = eattr[(size_t)e*8 + j];
    }
    wave_lds_fence();

    // ---- KPRE = Ltile[16x112] @ TW[112x128]  (M = 16 edges) ----
    for (int nt = 0; nt < 8; ++nt) {
      v8f acc = v8zero();
      #pragma unroll
      for (int kk = 0; kk < LDIM/4; ++kk) {
        int k0 = kk*4 + 2*hi;
        v2f a, b;
        a.x = sLK[nl*LK_LD + k0];
        a.y = sLK[nl*LK_LD + k0 + 1];
        b.x = sTW[k0*TW_LD + nt*16 + nl];
        b.y = sTW[(k0+1)*TW_LD + nt*16 + nl];
        acc = wmma4(a, b, acc);
      }
      #pragma unroll
      for (int r = 0; r < 8; ++r)
        sKP[(hi*8 + r)*KP_LD + nt*16 + nl] = acc[r];
    }
    wave_lds_fence();

    // ---- per-edge pipeline ----
    for (int ee = 0; ee < 16; ++ee) {
      const int eg = e0 + ee;
      const int dn = sDst[ee];

      float qv[Ct], ekv[Ct];
      #pragma unroll
      for (int t = 0; t < Ct; ++t) {
        int c = t*16 + nl;
        qv[t] = q[(size_t)dn*C + c];
        float s = bkedge[c];
        #pragma unroll
        for (int j = 0; j < 8; ++j) s = fmaf(sEA[ee*8 + j], Wkedge[c*8 + j], s);
        ekv[t] = s;
      }

      // k_labels = (kpre + bi2k)[16k x 8m] @ We2o^T[8m x C]
      const float bi = bi2k[nl];
      v8f kl[Ct];
      #pragma unroll
      for (int ct = 0; ct < Ct; ++ct) {
        v8f acc = v8zero();
        #pragma unroll
        for (int kk = 0; kk < 2; ++kk) {
          int m0 = kk*4 + 2*hi;
          v2f a, b;
          a.x = sKP[ee*KP_LD + m0*16 + nl] + bi;
          a.y = sKP[ee*KP_LD + (m0+1)*16 + nl] + bi;
          b.x = We2o[(ct*16 + nl)*8 + m0];
          b.y = We2o[(ct*16 + nl)*8 + m0 + 1];
          acc = wmma4(a, b, acc);
        }
        float be = be2o[ct*16 + nl];
        #pragma unroll
        for (int r = 0; r < 8; ++r) acc[r] += be;
        kl[ct] = acc;
      }

      // stage k_labels (D-layout -> row-major LDS) for A-fragments of k_key
      wave_lds_fence();
      #pragma unroll
      for (int ct = 0; ct < Ct; ++ct) {
        #pragma unroll
        for (int r = 0; r < 8; ++r)
          sLK[(hi*8 + r)*LK_LD + ct*16 + nl] = kl[ct][r];
      }
      wave_lds_fence();

      // k_key = k_labels[16k x C] @ Wkkey^T[C x C], fused into x partials
      float xr[8] = {0.f,0.f,0.f,0.f,0.f,0.f,0.f,0.f};
      for (int dt = 0; dt < Ct; ++dt) {
        v8f acc = v8zero();
        #pragma unroll
        for (int kk = 0; kk < C/4; ++kk) {
          int k0 = kk*4 + 2*hi;
          v2f a, b;
          a.x = sLK[nl*LK_LD + k0];
          a.y = sLK[nl*LK_LD + k0 + 1];
          b.x = sKK[k0*KK_LD + dt*16 + nl];
          b.y = sKK[(k0+1)*KK_LD + dt*16 + nl];
          acc = wmma4(a, b, acc);
        }
        const float bk = bkkey[dt*16 + nl];
        const float qt = qv[dt], et = ekv[dt];
        #pragma unroll
        for (int r = 0; r < 8; ++r) xr[r] = fmaf(qt, et + bk + acc[r], xr[r]);
      }

      // reduce over c (16-lane butterfly); lane holds x[k= hi*8+r]
      #pragma unroll
      for (int r = 0; r < 8; ++r) {
        float v = xr[r];
        v += __shfl_xor(v, 1, 32);
        v += __shfl_xor(v, 2, 32);
        v += __shfl_xor(v, 4, 32);
        v += __shfl_xor(v, 8, 32);
        xr[r] = v * 0.25f;                   // / sqrt(LABEL_K)
      }
      // softmax over 16 k
      float mx = xr[0];
      #pragma unroll
      for (int r = 1; r < 8; ++r) mx = fmaxf(mx, xr[r]);
      mx = fmaxf(mx, __shfl_xor(mx, 16, 32));
      float sum = 0.f;
      #pragma unroll
      for (int r = 0; r < 8; ++r) { xr[r] = __expf(xr[r] - mx); sum += xr[r]; }
      sum += __shfl_xor(sum, 16, 32);
      float rs = 1.f / sum;
      #pragma unroll
      for (int r = 0; r < 8; ++r) xr[r] *= rs;

      // out[c] = sum_k alpha[k] * k_labels[k][c]
      float o[Ct];
      #pragma unroll
      for (int t = 0; t < Ct; ++t) {
        float s = 0.f;
        #pragma unroll
        for (int r = 0; r < 8; ++r) s = fmaf(xr[r], kl[t][r], s);
        s += __shfl_xor(s, 16, 32);
        o[t] = s;
      }

      // GAT score a = Walpha . [q_e, out] + b, masked
      float ap = 0.f;
      if (hi == 0) {
        #pragma unroll
        for (int t = 0; t < Ct; ++t) {
          int c = t*16 + nl;
          ap = fmaf(Walpha[c], qv[t], ap);
          ap = fmaf(Walpha[C + c], o[t], ap);
        }
      }
      ap += __shfl_xor(ap, 1, 32);
      ap += __shfl_xor(ap, 2, 32);
      ap += __shfl_xor(ap, 4, 32);
      ap += __shfl_xor(ap, 8, 32);
      ap += __shfl_xor(ap, 16, 32);
      float aval = (ap + balpha[0]) * (sMsk[ee] * -1e6f + 1.0f);

      if (eg < E) {
        if (hi == 0) {
          #pragma unroll
          for (int t = 0; t < Ct; ++t)
            out_edge[(size_t)eg*C + t*16 + nl] = o[t];
        }
        if (lane == 0) {
          aedge[eg] = aval;
          unsigned int bits = __float_as_uint(aval);
          unsigned int key = (bits & 0x80000000u) ? ~bits : (bits | 0x80000000u);
          atomicMax(amaxenc + dn, key);   // monotonic float-order encoding
        }
      }
    }
  }
}

// ---------------------------------------------------------------------------
// Simple kernels
// ---------------------------------------------------------------------------
__global__ void lin_kernel(const float* __restrict__ x, const float* __restrict__ W,
                           const float* __restrict__ b, float* __restrict__ out,
                           int N, int Cin, int Cout) {
  long idx = (long)blockIdx.x*blockDim.x + threadIdx.x;
  if (idx >= (long)N*Cout) return;
  int n = (int)(idx / Cout), c = (int)(idx % Cout);
  const float* xr = x + (size_t)n*Cin;
  const float* wr = W + (size_t)c*Cin;
  float s = b[c];
  for (int i = 0; i < Cin; ++i) s = fmaf(xr[i], wr[i], s);
  out[idx] = s;
}

__global__ void zero_kernel(float* __restrict__ p, long n) {
  long i = (long)blockIdx.x*blockDim.x + threadIdx.x;
  if (i < n) p[i] = 0.f;
}

__global__ void pass1_kernel(const int* __restrict__ ei,
                             const unsigned int* __restrict__ amaxenc,
                             float* __restrict__ aedge, float* __restrict__ den, int E) {
  int e = blockIdx.x*blockDim.x + threadIdx.x;
  if (e >= E) return;
  int d = ei[E + e];
  unsigned int key = amaxenc[d];
  unsigned int bits = (key & 0x80000000u) ? (key ^ 0x80000000u) : ~key;
  float amax = __uint_as_float(bits);
  float ex = __expf(aedge[e] - amax);
  aedge[e] = ex;
  atomicAdd(den + d, ex);
}

__global__ void pass2_kernel(const int* __restrict__ ei, const float* __restrict__ aedge,
                             const float* __restrict__ out_edge, float* __restrict__ agg,
                             int E, int C) {
  long idx = (long)blockIdx.x*blockDim.x + threadIdx.x;
  if (idx >= (long)E*C) return;
  int e = (int)(idx / C), c = (int)(idx % C);
  int d = ei[E + e];
  atomicAdd(agg + (size_t)d*C + c, out_edge[idx] * aedge[e]);
}

__global__ void comb_kernel(const float* __restrict__ skip, const float* __restrict__ agg,
                            const float* __restrict__ den,
                            const float* __restrict__ W, const float* __restrict__ b,
                            float* __restrict__ out, int N, int C, int relu) {
  long idx = (long)blockIdx.x*blockDim.x + threadIdx.x;
  if (idx >= (long)N*C) return;
  int n = (int)(idx / C), o = (int)(idx % C);
  float rden = 1.0f / (den[n] + 1e-16f);
  const float* wr = W + (size_t)o*2*C;
  const float* sr = skip + (size_t)n*C;
  const float* ar = agg  + (size_t)n*C;
  float s = b[o];
  for (int c = 0; c < C; ++c) s = fmaf(wr[c], sr[c], s);
  for (int c = 0; c < C; ++c) s = fmaf(wr[C + c], ar[c]*rden, s);
  if (relu) s = fmaxf(s, 0.f);
  out[idx] = s;
}

// ---------------------------------------------------------------------------
extern "C" void kernel_launch(void* const* d_in, const int* in_sizes, int n_in,
                              void* d_out, int out_size, void* d_ws, size_t ws_size,
                              hipStream_t stream) {
  const int N = in_sizes[0] / 8;     // features [N,8]
  const int E = in_sizes[1] / 2;     // edge_index [2,E]

  const float* feats = (const float*)d_in[0];
  const int*   ei    = (const int*)d_in[1];
  const float* eattr = (const float*)d_in[2];
  const float* yv    = (const float*)d_in[3];
  const int*   emask = (const int*)d_in[4];
  const float* table = (const float*)d_in[5];
  auto P = [&](int L, int i) { return (const float*)d_in[6 + L*16 + i]; };

  float* ws = (float*)d_ws;
  const size_t NC = (size_t)N * 112;
  float* h0       = ws; ws += NC;
  float* h1       = ws; ws += NC;
  float* qbuf     = ws; ws += NC;
  float* skipbuf  = ws; ws += NC;
  float* scat     = ws; ws += (size_t)N * 114;   // [amaxenc N][den N][agg N*112]
  float* aedge    = ws; ws += (size_t)E;
  float* out_edge = ws; ws += (size_t)E * 112;
  unsigned int* amaxenc = (unsigned int*)scat;
  float* den = scat + N;
  float* agg = scat + 2*(size_t)N;

  const int Cs[3]  = {64, 64, 112};
  const int Ins[3] = {8, 64, 64};
  float* houts[3]  = {h0, h1, (float*)d_out};
  const float* hin = feats;

  for (int L = 0; L < 3; ++L) {
    const int C = Cs[L], IN = Ins[L];
    // node linears: q, skip
    {
      long tq = (long)N * C;
      int blocks = (int)((tq + 255) / 256);
      lin_kernel<<<blocks, 256, 0, stream>>>(hin, P(L,0), P(L,1), qbuf,    N, IN, C);
      lin_kernel<<<blocks, 256, 0, stream>>>(hin, P(L,2), P(L,3), skipbuf, N, IN, C);
    }
    // zero amax/den/agg
    {
      long nz = (long)N * 114;
      zero_kernel<<<(int)((nz + 255) / 256), 256, 0, stream>>>(scat, nz);
    }
    // heavy edge kernel (WMMA)
    {
      const size_t perWave = (size_t)(16*116 + 16*8 + 16*132 + 48);
      const size_t smemf = (size_t)112*136 + (size_t)C*(C + 8) + 4*perWave;
      const size_t smemb = smemf * sizeof(float);
      if (C == 64) {
        edge_kernel<64><<<1024, 128, smemb, stream>>>(
            yv, ei, eattr, emask, qbuf, table,
            P(L,8), P(L,9), P(L,6), P(L,7), P(L,10), P(L,11),
            P(L,12), P(L,13), P(L,4), P(L,5),
            out_edge, aedge, amaxenc, E);
      } else {
        edge_kernel<112><<<1024, 128, smemb, stream>>>(
            yv, ei, eattr, emask, qbuf, table,
            P(L,8), P(L,9), P(L,6), P(L,7), P(L,10), P(L,11),
            P(L,12), P(L,13), P(L,4), P(L,5),
            out_edge, aedge, amaxenc, E);
      }
    }
    // segment softmax + scatter
    pass1_kernel<<<(E + 255) / 256, 256, 0, stream>>>(ei, amaxenc, aedge, den, E);
    {
      long t2 = (long)E * C;
      pass2_kernel<<<(int)((t2 + 255) / 256), 256, 0, stream>>>(ei, aedge, out_edge, agg, E, C);
    }
    // comb + (relu except last)
    {
      long tc = (long)N * C;
      comb_kernel<<<(int)((tc + 255) / 256), 256, 0, stream>>>(
          skipbuf, agg, den, P(L,14), P(L,15), houts[L], N, C, (L < 2) ? 1 : 0);
    }
    hin = houts[L];
  }
}